// GraphNN_43379169689656
// MI455X (gfx1250) — compile-verified
//
#include <hip/hip_runtime.h>

typedef float v2f __attribute__((ext_vector_type(2)));
typedef float v8f __attribute__((ext_vector_type(8)));

#define BSZ 256
#define NJ 64
#define NM 16
#define NNODE 80
#define NH 5
#define NNODES (BSZ * NNODE)   // 20480

__device__ __forceinline__ float lrelu(float x, float s) { return x > 0.f ? x : s * x; }

// ---------------------------------------------------------------------------
// Build node features, padded to 8 columns (col 7 = 0).
// jobs:     [h, L, 0, 0, 0, j+1, 0, 0]   machines: [0, 0, W, P, N, 0, m+1, 0]
// ---------------------------------------------------------------------------
__global__ void feat_kernel(const float* __restrict__ h, const float* __restrict__ L,
                            const float* __restrict__ W, const float* __restrict__ P,
                            const float* __restrict__ N, float* __restrict__ X) {
  int t = blockIdx.x * blockDim.x + threadIdx.x;
  if (t >= NNODES * 8) return;
  int g = t >> 3, col = t & 7;
  int b = g / NNODE, c = g % NNODE;
  float v = 0.f;
  if (c < NJ) {
    if (col == 0) v = h[b * NJ + c];
    else if (col == 1) v = L[b * NJ + c];
    else if (col == 5) v = (float)(c + 1);
  } else {
    int m = c - NJ;
    if (col == 2) v = W[b];
    else if (col == 3) v = P[b];
    else if (col == 4) v = N[b];
    else if (col == 6) v = (float)(m + 1);
  }
  X[t] = v;
}

// Zero-pad Wfc [din x ncols] -> Wp [din_pad x ncols] so GEMM needs no bounds.
__global__ void pad_w_kernel(const float* __restrict__ Wfc, float* __restrict__ Wp,
                             int din, int din_pad, int ncols) {
  int t = blockIdx.x * blockDim.x + threadIdx.x;
  if (t >= din_pad * ncols) return;
  int k = t / ncols;
  Wp[t] = (k < din) ? Wfc[t] : 0.f;
}

// ---------------------------------------------------------------------------
// FT[n, ncols] = X[n, DINP] @ Wp[DINP, ncols] using V_WMMA_F32_16X16X4_F32.
// Block = 128 threads = 4 waves; each wave owns one 16x16 tile (4 row tiles
// share one 16-col block of B, staged once in LDS). K loop fully unrolled
// at compile time (DINP/4 WMMAs back-to-back), no divergence -> EXEC all 1s.
// A layout (16x4 f32): lanes 0-15 hold {K=0,K=1} for row M=lane,
//                      lanes 16-31 hold {K=2,K=3} for row M=lane-16.
// B layout (4x16 f32): VGPR0 = K=0 (lanes 0-15) / K=2 (lanes 16-31),
//                      VGPR1 = K=1 / K=3, N = lane%16.
// C/D: VGPR v -> row base+v (lanes 0-15) / base+v+8 (lanes 16-31), N=lane%16.
// ---------------------------------------------------------------------------
template <int DINP>
__global__ void gemm_wmma_kernel(const float* __restrict__ X, const float* __restrict__ Wp,
                                 float* __restrict__ FT, int ncols) {
  __shared__ float Bs[DINP * 16];
  int lane = threadIdx.x & 31;
  int wave = threadIdx.x >> 5;                       // 0..3
  int colBase = blockIdx.x * 16;
  int rowBase = (blockIdx.y * 4 + wave) * 16;

  // Cooperative stage of B tile [DINP x 16] into LDS.
  for (int i = threadIdx.x; i < DINP * 16; i += 128) {
    int k = i >> 4, n = i & 15;
    Bs[i] = Wp[(size_t)k * ncols + colBase + n];
  }
  __syncthreads();

  int half = lane >> 4;
  int lm = lane & 15;
  v8f acc = {};
  const float* xrow = X + (size_t)(rowBase + lm) * DINP;   // 32B-aligned rows
#pragma unroll
  for (int kb = 0; kb < DINP; kb += 4) {
    int k0 = kb + 2 * half;                          // even -> 8B-aligned
    v2f a = *(const v2f*)(xrow + k0);
    v2f bm;
    bm.x = Bs[k0 * 16 + lm];
    bm.y = Bs[(k0 + 1) * 16 + lm];
    acc = __builtin_amdgcn_wmma_f32_16x16x4_f32(false, a, false, bm, (short)0, acc,
                                                false, false);
  }
#pragma unroll
  for (int v = 0; v < 8; ++v)
    FT[(size_t)(rowBase + v + 8 * half) * ncols + colBase + lm] = acc[v];
}

// ---------------------------------------------------------------------------
// el[n,h] = <ft[n,h,:], al[h,:]>,  er[n,h] = <ft[n,h,:], ar[h,:]>
// ---------------------------------------------------------------------------
__global__ void elr_kernel(const float* __restrict__ FT, const float* __restrict__ al,
                           const float* __restrict__ ar, float* __restrict__ EL,
                           float* __restrict__ ER, int dout, int ncols) {
  int t = blockIdx.x * blockDim.x + threadIdx.x;
  if (t >= NNODES * NH) return;
  int n = t / NH, h = t % NH;
  const float* f = FT + (size_t)n * ncols + h * dout;
  float sl = 0.f, sr = 0.f;
  for (int d = 0; d < dout; ++d) {
    float v = f[d];
    sl += v * al[h * dout + d];
    sr += v * ar[h * dout + d];
  }
  EL[t] = sl;
  ER[t] = sr;
}

// ce[h] = sum_d We[h*dout+d] * ae[h*dout+d]   (ee = ef * ce[h])
__global__ void ce_kernel(const float* __restrict__ We, const float* __restrict__ ae,
                          float* __restrict__ CE, int dout) {
  int h = threadIdx.x;
  if (h >= NH) return;
  float s = 0.f;
  for (int d = 0; d < dout; ++d) s += We[h * dout + d] * ae[h * dout + d];
  CE[h] = s;
}

// ---------------------------------------------------------------------------
// Dense per-dst edge-softmax + aggregation + bias + leaky(0.01) + head mean.
// Thread per (dst node g, channel d). Job dsts: 4 fixed srcs (j-1..j-4 mod J)
// with edge feature T[b,r,c]; machine dsts: all 64 jobs, ef = 0.
// ---------------------------------------------------------------------------
__global__ void agg_kernel(const float* __restrict__ FT, const float* __restrict__ EL,
                           const float* __restrict__ ER, const float* __restrict__ CE,
                           const float* __restrict__ T, const float* __restrict__ bias,
                           float* __restrict__ OUT, int dout, int ncols, int out_stride) {
  int t = blockIdx.x * blockDim.x + threadIdx.x;
  if (t >= NNODES * dout) return;
  int g = t / dout, d = t % dout;
  int b = g / NNODE, c = g % NNODE;
  float sumh = 0.f;
  for (int h = 0; h < NH; ++h) {
    float erd = ER[(size_t)g * NH + h];
    float ceh = CE[h];
    float mx = -1e30f, den = 0.f, num = 0.f;
    if (c < NJ) {
#pragma unroll
      for (int k = 1; k <= 4; ++k) {
        int r = c - k; if (r < 0) r += NJ;
        float ef = T[((size_t)b * NJ + r) * NJ + c];
        float e = lrelu(EL[(size_t)(b * NNODE + r) * NH + h] + erd + ef * ceh, 0.2f);
        mx = fmaxf(mx, e);
      }
#pragma unroll
      for (int k = 1; k <= 4; ++k) {
        int r = c - k; if (r < 0) r += NJ;
        float ef = T[((size_t)b * NJ + r) * NJ + c];
        float e = lrelu(EL[(size_t)(b * NNODE + r) * NH + h] + erd + ef * ceh, 0.2f);
        float w = __expf(e - mx);
        den += w;
        num += w * FT[(size_t)(b * NNODE + r) * ncols + h * dout + d];
      }
    } else {
      for (int r = 0; r < NJ; ++r) {
        float e = lrelu(EL[(size_t)(b * NNODE + r) * NH + h] + erd, 0.2f);
        mx = fmaxf(mx, e);
      }
      for (int r = 0; r < NJ; ++r) {
        float e = lrelu(EL[(size_t)(b * NNODE + r) * NH + h] + erd, 0.2f);
        float w = __expf(e - mx);
        den += w;
        num += w * FT[(size_t)(b * NNODE + r) * ncols + h * dout + d];
      }
    }
    float o = num / den + bias[h * dout + d];
    sumh += lrelu(o, 0.01f);
  }
  OUT[(size_t)g * out_stride + d] = sumh * (1.0f / NH);
}

// ---------------------------------------------------------------------------
extern "C" void kernel_launch(void* const* d_in, const int* in_sizes, int n_in,
                              void* d_out, int out_size, void* d_ws, size_t ws_size,
                              hipStream_t stream) {
  const float* hI = (const float*)d_in[0];
  const float* LI = (const float*)d_in[1];
  const float* WI = (const float*)d_in[2];
  const float* PI = (const float*)d_in[3];
  const float* NI = (const float*)d_in[4];
  const float* TI = (const float*)d_in[5];
  // params flattened in insertion order: per layer {Wfc, We, al, ar, ae, b}
  const float* prm[18];
  for (int i = 0; i < 18; ++i) prm[i] = (const float*)d_in[6 + i];

  float* ws = (float*)d_ws;
  float* x0 = ws;                               // 20480*64
  float* x1 = x0 + (size_t)NNODES * 64;         // 20480*64
  float* ft = x1 + (size_t)NNODES * 64;         // 20480*320
  float* el = ft + (size_t)NNODES * 320;        // 20480*5
  float* er = el + (size_t)NNODES * NH;         // 20480*5
  float* ce = er + (size_t)NNODES * NH;         // 8
  float* wp = ce + 8;                           // 64*320 padded weights

  feat_kernel<<<(NNODES * 8 + 255) / 256, 256, 0, stream>>>(hI, LI, WI, PI, NI, x0);

  const int din[3]   = {7, 16, 64};
  const int dinp[3]  = {8, 16, 64};
  const int doutA[3] = {16, 64, 64};
  const int ncolsA[3] = {80, 320, 320};
  const int ostr[3]  = {16, 64, 64};

  float* xin = x0;
  float* xout = x1;
  for (int li = 0; li < 3; ++li) {
    const float* Wfc = prm[6 * li + 0];
    const float* We  = prm[6 * li + 1];
    const float* al  = prm[6 * li + 2];
    const float* ar  = prm[6 * li + 3];
    const float* ae  = prm[6 * li + 4];
    const float* bb  = prm[6 * li + 5];
    int dc = doutA[li], nc = ncolsA[li];

    int padN = dinp[li] * nc;
    pad_w_kernel<<<(padN + 255) / 256, 256, 0, stream>>>(Wfc, wp, din[li], dinp[li], nc);

    dim3 gg(nc / 16, NNODES / 16 / 4);   // 4 waves per block, 4 row tiles
    if (li == 0)
      gemm_wmma_kernel<8><<<gg, 128, 0, stream>>>(xin, wp, ft, nc);
    else if (li == 1)
      gemm_wmma_kernel<16><<<gg, 128, 0, stream>>>(xin, wp, ft, nc);
    else
      gemm_wmma_kernel<64><<<gg, 128, 0, stream>>>(xin, wp, ft, nc);

    elr_kernel<<<(NNODES * NH + 255) / 256, 256, 0, stream>>>(ft, al, ar, el, er, dc, nc);
    ce_kernel<<<1, 32, 0, stream>>>(We, ae, ce, dc);

    float* out = (li == 2) ? (float*)d_out : xout;
    int tot = NNODES * dc;
    agg_kernel<<<(tot + 255) / 256, 256, 0, stream>>>(ft, el, er, ce, TI, bb, out,
                                                      dc, nc, ostr[li]);
    float* tmp = xin; xin = xout; xout = tmp;
  }
}